// FreeYOLOv2_53824530153482
// MI455X (gfx1250) — compile-verified
//
#include <hip/hip_runtime.h>
#include <stdint.h>

#define NCLS   80
#define TOPKN  1000
#define NBINS  4096
#define CAP    4096
#define NTOT   3000
#define NWORD  94           // ceil(3000/32)
#define CONF   0.05f
#define NMS_T  0.6f
#define CLS_OFF 8192.0f

typedef float v2f __attribute__((ext_vector_type(2)));
typedef float v8f __attribute__((ext_vector_type(8)));

__device__ __forceinline__ unsigned ordkey(float f) {
    unsigned b = __float_as_uint(f);
    return (b & 0x80000000u) ? ~b : (b | 0x80000000u);
}

// ---------------- Pass 1: per-level histogram of logit order-keys -------------
__global__ void hist_kernel(const float* __restrict__ x, unsigned n,
                            unsigned* __restrict__ gh) {
    __shared__ unsigned lh[NBINS];
    for (int i = threadIdx.x; i < NBINS; i += blockDim.x) lh[i] = 0;
    __syncthreads();
    unsigned stride = blockDim.x * gridDim.x;
    for (unsigned i = blockIdx.x * blockDim.x + threadIdx.x; i < n; i += stride)
        atomicAdd(&lh[ordkey(x[i]) >> 20], 1u);
    __syncthreads();
    for (int i = threadIdx.x; i < NBINS; i += blockDim.x) {
        unsigned v = lh[i];
        if (v) atomicAdd(&gh[i], v);
    }
}

// ---------------- Pass 2: find threshold bin (suffix count >= 1000) ----------
__global__ void thresh_kernel(const unsigned* __restrict__ hist,
                              unsigned* __restrict__ thr) {
    if (threadIdx.x) return;
    const unsigned* h = hist + blockIdx.x * NBINS;
    unsigned acc = 0;
    int b;
    for (b = NBINS - 1; b > 0; --b) { acc += h[b]; if (acc >= TOPKN) break; }
    thr[blockIdx.x] = (unsigned)b;
}

// ---------------- Pass 3: compact candidates above threshold bin -------------
__global__ void compact_kernel(const float* __restrict__ x, unsigned n, unsigned M,
                               const unsigned* __restrict__ thr, int lev,
                               unsigned* __restrict__ cnt, uint64_t* __restrict__ cand) {
    unsigned tb = thr[lev];
    unsigned stride = blockDim.x * gridDim.x;
    for (unsigned i = blockIdx.x * blockDim.x + threadIdx.x; i < n; i += stride) {
        unsigned u = ordkey(x[i]);
        if ((u >> 20) >= tb) {
            unsigned m = i % M, c = i / M;              // layout [C][M]
            unsigned flat = m * NCLS + c;               // reference flat = m*C + c
            unsigned pos = atomicAdd(&cnt[lev], 1u);
            if (pos < CAP)
                cand[(size_t)lev * CAP + pos] =
                    ~(((uint64_t)u << 32) | (uint64_t)(0xFFFFFFFFu - flat));
        }
    }
}

__device__ __forceinline__ void bitonic_sort_shared(uint64_t* sk, int n) {
    for (unsigned k = 2; k <= (unsigned)n; k <<= 1)
        for (unsigned j = k >> 1; j > 0; j >>= 1) {
            for (unsigned i = threadIdx.x; i < (unsigned)n; i += blockDim.x) {
                unsigned ixj = i ^ j;
                if (ixj > i) {
                    uint64_t a = sk[i], b = sk[ixj];
                    bool up = ((i & k) == 0);
                    if (up ? (a > b) : (a < b)) { sk[i] = b; sk[ixj] = a; }
                }
            }
            __syncthreads();
        }
}

// ---------------- Pass 4: per-level top-1000 via bitonic sort ----------------
__global__ __launch_bounds__(1024)
void topk_kernel(const uint64_t* __restrict__ cand, const unsigned* __restrict__ cnt,
                 int* __restrict__ canchor, int* __restrict__ clabel,
                 float* __restrict__ cscore, uint64_t* __restrict__ ckey) {
    __shared__ uint64_t sk[CAP];
    int lev = blockIdx.x;
    unsigned n = cnt[lev]; if (n > CAP) n = CAP;
    for (int i = threadIdx.x; i < CAP; i += blockDim.x)
        sk[i] = (i < (int)n) ? cand[(size_t)lev * CAP + i] : ~0ull;
    __syncthreads();
    bitonic_sort_shared(sk, CAP);   // ascending on inverted key == descending score
    for (int r = threadIdx.x; r < TOPKN; r += blockDim.x) {
        uint64_t key = ~sk[r];
        unsigned u = (unsigned)(key >> 32);
        unsigned flat = 0xFFFFFFFFu - (unsigned)(key & 0xFFFFFFFFu);
        unsigned bb = (u & 0x80000000u) ? (u ^ 0x80000000u) : ~u;  // un-orderkey
        float lg = __uint_as_float(bb);
        float s = 1.0f / (1.0f + __expf(-lg));
        float ts = (s > CONF) ? s : 0.0f;
        int pos = lev * TOPKN + r;
        canchor[pos] = (int)(flat / NCLS);
        clabel[pos]  = (int)(flat % NCLS);
        cscore[pos]  = ts;
        // final sort key: descending thresholded score, stable on concat index
        ckey[pos] = ((uint64_t)ordkey(ts) << 32) | (uint64_t)(0xFFFFFFFFu - (unsigned)pos);
    }
}

// ---------------- Pass 5: WMMA DFL softmax-expectation box decode ------------
// D = A(16x4: rows all = proj_w chunk) x B(4x16: col j = softmax probs of entry j)
// => every lane gets dist[entry = lane%16] in acc[0]. 4 K-chunks accumulate K=16.
__global__ void decode_kernel(const float* __restrict__ r3, const float* __restrict__ r4,
                              const float* __restrict__ r5, const float* __restrict__ proj,
                              const int* __restrict__ canchor, const int* __restrict__ clabel,
                              float* __restrict__ cbox, float* __restrict__ sbox,
                              float* __restrict__ areao) {
    int lev  = blockIdx.x >> 3;
    int wave = (blockIdx.x & 7) * (blockDim.x >> 5) + (threadIdx.x >> 5);
    if (wave >= (TOPKN + 15) / 16) return;      // 63 groups of 16 entries
    int lane = threadIdx.x & 31;
    int col = lane & 15, half = lane >> 4;
    int e = wave * 16 + col;
    bool valid = e < TOPKN;
    int pos = lev * TOPKN + (valid ? e : (TOPKN - 1));
    unsigned m = (unsigned)canchor[pos];
    if (!valid) m = 0;

    const float* reg; unsigned M, sh; float st;
    if (lev == 0)      { reg = r3; M = 512u*512u; sh = 9; st = 8.f;  }
    else if (lev == 1) { reg = r4; M = 256u*256u; sh = 8; st = 16.f; }
    else               { reg = r5; M = 128u*128u; sh = 7; st = 32.f; }

    // A operand: proj_w chunk values; half-wave K split matches 16x4 A layout
    v2f av[4];
#pragma unroll
    for (int cc = 0; cc < 4; cc++) {
        int r0 = cc * 4 + half * 2;
        v2f t = { proj[r0], proj[r0 + 1] };
        av[cc] = t;
    }

    float d[4];
#pragma unroll
    for (int f = 0; f < 4; f++) {
        float xv[8];
#pragma unroll
        for (int cc = 0; cc < 4; cc++) {
            int ch = f * 16 + cc * 4 + half * 2;
            xv[2*cc]     = reg[(size_t)ch * M + m];
            xv[2*cc + 1] = reg[(size_t)(ch + 1) * M + m];
        }
        // row softmax: each bin-row is split across lanes L and L+16
        float mx = xv[0];
#pragma unroll
        for (int t = 1; t < 8; t++) mx = fmaxf(mx, xv[t]);
        mx = fmaxf(mx, __shfl_xor(mx, 16));
        float ev[8], sum = 0.f;
#pragma unroll
        for (int t = 0; t < 8; t++) { ev[t] = __expf(xv[t] - mx); sum += ev[t]; }
        sum += __shfl_xor(sum, 16);
        float inv = 1.0f / sum;

        v8f acc = {0.f, 0.f, 0.f, 0.f, 0.f, 0.f, 0.f, 0.f};
#pragma unroll
        for (int cc = 0; cc < 4; cc++) {
            v2f bv = { ev[2*cc] * inv, ev[2*cc + 1] * inv };
            acc = __builtin_amdgcn_wmma_f32_16x16x4_f32(
                false, av[cc], false, bv, (short)0, acc, false, false);
        }
        d[f] = acc[0];
    }

    if (valid && half == 0) {
        float ax = ((float)(m & ((1u << sh) - 1u)) + 0.5f) * st;
        float ay = ((float)(m >> sh) + 0.5f) * st;
        float x1 = ax - d[0] * st, y1 = ay - d[1] * st;
        float x2 = ax + d[2] * st, y2 = ay + d[3] * st;
        ((float4*)cbox)[pos] = make_float4(x1, y1, x2, y2);
        float off = (float)clabel[pos] * CLS_OFF;
        ((float4*)sbox)[pos] = make_float4(x1 + off, y1 + off, x2 + off, y2 + off);
        areao[pos] = (x2 - x1) * (y2 - y1);
    }
}

// ---------------- Pass 6: global stable sort of 3000, emit boxes/labels ------
__global__ __launch_bounds__(1024)
void final_sort_kernel(const uint64_t* __restrict__ ckey, const float* __restrict__ cbox,
                       const int* __restrict__ clabel, const float* __restrict__ cscore,
                       const float* __restrict__ sbox, const float* __restrict__ area,
                       float* __restrict__ out, float* __restrict__ sscore,
                       float* __restrict__ ssbox, float* __restrict__ sarea,
                       unsigned* __restrict__ keepinit) {
    __shared__ uint64_t sk[4096];
    __shared__ float ss[NTOT];
    for (int i = threadIdx.x; i < 4096; i += blockDim.x)
        sk[i] = (i < NTOT) ? ~ckey[i] : ~0ull;
    __syncthreads();
    bitonic_sort_shared(sk, 4096);
    const float4* cb4 = (const float4*)cbox;
    const float4* sb4 = (const float4*)sbox;
    float4* ob4 = (float4*)out;
    float4* ssb4 = (float4*)ssbox;
    for (int r = threadIdx.x; r < NTOT; r += blockDim.x) {
        uint64_t key = ~sk[r];
        unsigned pos = 0xFFFFFFFFu - (unsigned)(key & 0xFFFFFFFFu);
        ob4[r] = cb4[pos];                            // bboxes out [0 .. 12000)
        out[5 * NTOT + r] = (float)clabel[pos];       // labels out [15000 .. 18000)
        float sc = cscore[pos];
        ss[r] = sc; sscore[r] = sc;
        ssb4[r] = sb4[pos];
        sarea[r] = area[pos];
    }
    __syncthreads();
    for (int w = threadIdx.x; w < 128; w += blockDim.x) {
        unsigned bits = 0;
        if (w < NWORD)
            for (int b = 0; b < 32; b++) {
                int j = w * 32 + b;
                if (j < NTOT && ss[j] > 0.0f) bits |= 1u << b;
            }
        keepinit[w] = bits;
    }
}

// ---------------- Pass 7: parallel suppression bit-matrix --------------------
__global__ void supmat_kernel(const float* __restrict__ ssbox,
                              const float* __restrict__ sarea,
                              unsigned* __restrict__ sup) {
    int gid = blockIdx.x * blockDim.x + threadIdx.x;
    if (gid >= NTOT * NWORD) return;
    int i = gid / NWORD, w = gid - i * NWORD;
    const float4* b4 = (const float4*)ssbox;
    float4 bi = b4[i]; float ai = sarea[i];
    unsigned bits = 0;
    int j0 = w * 32;
    for (int b = 0; b < 32; b++) {
        int j = j0 + b;
        if (j > i && j < NTOT) {
            float4 bj = b4[j];
            float iw = fmaxf(fminf(bi.z, bj.z) - fmaxf(bi.x, bj.x), 0.f);
            float ih = fmaxf(fminf(bi.w, bj.w) - fmaxf(bi.y, bj.y), 0.f);
            float inter = iw * ih;
            float iou = inter / (ai + sarea[j] - inter + 1e-9f);
            if (iou > NMS_T) bits |= 1u << b;
        }
    }
    sup[gid] = bits;
}

// ---------------- Pass 8: single-wave barrier-free greedy sweep --------------
__global__ void sweep_kernel(const unsigned* __restrict__ sup,
                             const unsigned* __restrict__ keepinit,
                             const float* __restrict__ sscore,
                             float* __restrict__ out) {
    int l = threadIdx.x;                 // lane l owns words l, l+32, l+64
    unsigned k0 = keepinit[l];
    unsigned k1 = keepinit[l + 32];
    unsigned k2 = keepinit[l + 64];      // words >= 94 are zero
    unsigned a0 = sup[l];
    unsigned a1 = sup[l + 32];
    unsigned a2 = (l + 64 < NWORD) ? sup[l + 64] : 0u;
    for (int i = 0; i < NTOT; i++) {
        unsigned n0 = 0, n1 = 0, n2 = 0;             // prefetch next row
        if (i + 1 < NTOT) {
            size_t ro = (size_t)(i + 1) * NWORD;
            n0 = sup[ro + l];
            n1 = sup[ro + l + 32];
            n2 = (l + 64 < NWORD) ? sup[ro + l + 64] : 0u;
        }
        int w = i >> 5, r = w >> 5, src = w & 31;
        unsigned kw = (r == 0) ? k0 : (r == 1) ? k1 : k2;
        unsigned word = (unsigned)__shfl((int)kw, src);
        if ((word >> (i & 31)) & 1u) { k0 &= ~a0; k1 &= ~a1; k2 &= ~a2; }
        a0 = n0; a1 = n1; a2 = n2;
    }
    // emit scores*keep and keep
    for (int jb = 0; jb < NWORD * 32; jb += 32) {
        int w = jb >> 5, r = w >> 5, src = w & 31;
        unsigned w0 = (unsigned)__shfl((int)k0, src);
        unsigned w1 = (unsigned)__shfl((int)k1, src);
        unsigned w2 = (unsigned)__shfl((int)k2, src);
        unsigned word = (r == 0) ? w0 : (r == 1) ? w1 : w2;
        int j = jb + l;
        if (j < NTOT) {
            float bit = (float)((word >> l) & 1u);
            out[4 * NTOT + j] = sscore[j] * bit;     // scores [12000 .. 15000)
            out[6 * NTOT + j] = bit;                 // keep   [18000 .. 21000)
        }
    }
}

extern "C" void kernel_launch(void* const* d_in, const int* in_sizes, int n_in,
                              void* d_out, int out_size, void* d_ws, size_t ws_size,
                              hipStream_t stream) {
    (void)in_sizes; (void)n_in; (void)out_size; (void)ws_size;
    const float* cls[3]  = {(const float*)d_in[0], (const float*)d_in[2], (const float*)d_in[4]};
    const float* regp[3] = {(const float*)d_in[1], (const float*)d_in[3], (const float*)d_in[5]};
    const float* proj = (const float*)d_in[6];
    float* out = (float*)d_out;

    uint8_t* w8 = (uint8_t*)d_ws;
    unsigned* hist = (unsigned*)w8;            // 3*4096 u32
    unsigned* cnt  = hist + 3 * NBINS;         // 3 u32
    unsigned* thr  = cnt + 3;                  // 3 u32 (+pad) -> 49184 B ctrl
    uint64_t* cand    = (uint64_t*)(w8 + 49184);    // 3*4096 u64
    int*      canchor = (int*)     (w8 + 147488);
    int*      clabel  = (int*)     (w8 + 159488);
    float*    cscore  = (float*)   (w8 + 171488);
    uint64_t* ckey    = (uint64_t*)(w8 + 183488);
    float*    cbox    = (float*)   (w8 + 207488);
    float*    sbox    = (float*)   (w8 + 255488);
    float*    area    = (float*)   (w8 + 303488);
    float*    sscore  = (float*)   (w8 + 315488);
    float*    ssbox   = (float*)   (w8 + 327488);
    float*    sarea   = (float*)   (w8 + 375488);
    unsigned* keepinit= (unsigned*)(w8 + 387488);
    unsigned* sup     = (unsigned*)(w8 + 388000);   // 3000*94 u32

    hipMemsetAsync(d_ws, 0, 49184, stream);         // hist + counters + thresh

    unsigned nvals[3] = {NCLS * 512u * 512u, NCLS * 256u * 256u, NCLS * 128u * 128u};
    unsigned Ms[3]    = {512u * 512u, 256u * 256u, 128u * 128u};

    for (int lev = 0; lev < 3; lev++)
        hist_kernel<<<1024, 256, 0, stream>>>(cls[lev], nvals[lev], hist + lev * NBINS);
    thresh_kernel<<<3, 32, 0, stream>>>(hist, thr);
    for (int lev = 0; lev < 3; lev++)
        compact_kernel<<<1024, 256, 0, stream>>>(cls[lev], nvals[lev], Ms[lev],
                                                 thr, lev, cnt, cand);
    topk_kernel<<<3, 1024, 0, stream>>>(cand, cnt, canchor, clabel, cscore, ckey);
    decode_kernel<<<24, 256, 0, stream>>>(regp[0], regp[1], regp[2], proj,
                                          canchor, clabel, cbox, sbox, area);
    final_sort_kernel<<<1, 1024, 0, stream>>>(ckey, cbox, clabel, cscore, sbox, area,
                                              out, sscore, ssbox, sarea, keepinit);
    supmat_kernel<<<(NTOT * NWORD + 255) / 256, 256, 0, stream>>>(ssbox, sarea, sup);
    sweep_kernel<<<1, 32, 0, stream>>>(sup, keepinit, sscore, out);
}